// GlobalEncoder_37211596653158
// MI455X (gfx1250) — compile-verified
//
#include <hip/hip_runtime.h>
#include <hip/hip_bf16.h>
#include <math.h>

typedef __attribute__((ext_vector_type(2))) float v2f;
typedef __attribute__((ext_vector_type(8))) float v8f;

// ---------------------------------------------------------------------------
// utility kernels
// ---------------------------------------------------------------------------
__global__ void fillf_k(float* p, float v, int n) {
  int i = blockIdx.x * blockDim.x + threadIdx.x;
  if (i < n) p[i] = v;
}
__global__ void filli_k(int* p, int v, int n) {
  int i = blockIdx.x * blockDim.x + threadIdx.x;
  if (i < n) p[i] = v;
}
__global__ void biasb_k(float* p, const float* __restrict__ b, int n, int M) {
  int i = blockIdx.x * blockDim.x + threadIdx.x;
  if (i < n) p[i] = b[i % M];
}
// z = mu + eps * exp(lv)   (used for atom_z and mol_z)
__global__ void reparam_k(const float* __restrict__ mu, const float* __restrict__ lv,
                          const float* __restrict__ eps, float* __restrict__ z, int n) {
  int i = blockIdx.x * blockDim.x + threadIdx.x;
  if (i < n) z[i] = mu[i] + eps[i] * __expf(lv[i]);
}
__global__ void segsum_k(const float* __restrict__ h, const int* __restrict__ batch,
                         float* __restrict__ out, int n) {  // n = N*512
  int i = blockIdx.x * blockDim.x + threadIdx.x;
  if (i < n) {
    int r = i >> 9, c = i & 511;
    atomicAdd(&out[(size_t)batch[r] * 512 + c], h[i]);
  }
}
// sum edge attrs (for self-loop mean) with LDS partial reduction
__global__ void easum_k(const float* __restrict__ ea, int E, float* __restrict__ sums) {
  __shared__ float s[13];
  if (threadIdx.x < 13) s[threadIdx.x] = 0.f;
  __syncthreads();
  int i = blockIdx.x * blockDim.x + threadIdx.x;
  if (i < E) {
#pragma unroll
    for (int k = 0; k < 13; ++k) atomicAdd(&s[k], ea[(size_t)i * 13 + k]);
  }
  __syncthreads();
  if (threadIdx.x < 13) atomicAdd(&sums[threadIdx.x], s[threadIdx.x]);
}
// type bucketing for hetero layers
__global__ void hist_k(const int* __restrict__ ty, int n, int* cnt) {
  int i = blockIdx.x * blockDim.x + threadIdx.x;
  if (i < n) atomicAdd(&cnt[ty[i]], 1);
}
__global__ void scan_k(const int* __restrict__ cnt, int* rstart, int* tstart) {
  int r = 0, tt = 0;
  for (int t = 0; t < 10; ++t) {
    rstart[t] = r; tstart[t] = tt;
    r += cnt[t]; tt += (cnt[t] + 15) >> 4;
  }
  rstart[10] = r; tstart[10] = tt;
}
__global__ void perm_k(const int* __restrict__ ty, int n, const int* __restrict__ rstart,
                       int* cur, int* perm) {
  int i = blockIdx.x * blockDim.x + threadIdx.x;
  if (i < n) {
    int t = ty[i];
    int s = rstart[t] + atomicAdd(&cur[t], 1);
    perm[s] = i;
  }
}

// ---------------------------------------------------------------------------
// WMMA micro-kernel: accumulate K-panel into 4 accumulators (16 rows x 64 cols)
// ap: A row pointer (pre-offset by row*ldx + klo), wp: W panel pointer
// (pre-offset by klo*M + cbase + rsub). Straight-line loop, pointer increments.
// ---------------------------------------------------------------------------
__device__ __forceinline__ void wmma_panel(const float* __restrict__ ap,
                                           const float* __restrict__ wp,
                                           int Ksteps, int M, v8f acc[4]) {
  for (int k = 0; k < Ksteps; ++k) {
    v2f a;
    a.x = ap[0];
    a.y = ap[1];
    const float* w1 = wp + M;
#pragma unroll
    for (int j = 0; j < 4; ++j) {
      v2f b;
      b.x = wp[j * 16];
      b.y = w1[j * 16];
      acc[j] = __builtin_amdgcn_wmma_f32_16x16x4_f32(false, a, false, b, (short)0,
                                                     acc[j], false, false);
    }
    ap += 4;
    wp += 4 * (size_t)M;
  }
}

// ---------------------------------------------------------------------------
// WMMA GEMM:  Y[rows,M] = act( concat(X[rows,K1], X2[rows,K2]) @ W[K1+K2,M] + bias )
// one wave -> 16 rows x 64 cols; 4 waves/block; act: 0=none 1=relu 2=min(.,10)
// ---------------------------------------------------------------------------
__global__ void __launch_bounds__(128)
gemm_k(const float* __restrict__ X, int K1, const float* __restrict__ X2, int K2,
       const float* __restrict__ W, const float* __restrict__ bias,
       float* __restrict__ Y, int nrows, int M, int act) {
  int lane = threadIdx.x & 31;
  int rbase = blockIdx.x * 64 + (threadIdx.x >> 5) * 16;
  if (rbase >= nrows) return;
  int cbase = blockIdx.y * 64;
  int rsub = lane & 15;
  int hi = lane >> 4;
  int klo = hi * 2;
  int row = rbase + rsub;
  int rl = row < nrows ? row : nrows - 1;
  v8f z = {};
  v8f acc[4] = {z, z, z, z};
  wmma_panel(X + (size_t)rl * K1 + klo,
             W + (size_t)klo * M + cbase + rsub, K1 >> 2, M, acc);
  if (K2 > 0)
    wmma_panel(X2 + (size_t)rl * K2 + klo,
               W + (size_t)(K1 + klo) * M + cbase + rsub, K2 >> 2, M, acc);
#pragma unroll
  for (int j = 0; j < 4; ++j) {
    int col = cbase + j * 16 + rsub;
    float bv = bias ? bias[col] : 0.f;
#pragma unroll
    for (int v = 0; v < 8; ++v) {
      int r = rbase + v + 8 * hi;
      if (r < nrows) {
        float o = acc[j][v] + bv;
        if (act == 1) o = o > 0.f ? o : 0.f;
        else if (act == 2) o = o < 10.f ? o : 10.f;
        Y[(size_t)r * M + col] = o;
      }
    }
  }
}

// ---------------------------------------------------------------------------
// Hetero (per-atom-type) WMMA GEMM over type-sorted row buckets.
// W: [10, K, M]  Bv: [10, M]. Rows gathered through perm, results scattered.
// ---------------------------------------------------------------------------
__global__ void __launch_bounds__(128)
het_k(const float* __restrict__ X, int K1, const float* __restrict__ X2, int K2,
      const float* __restrict__ W, const float* __restrict__ Bv,
      float* __restrict__ Y, int M, int act,
      const int* __restrict__ perm, const int* __restrict__ rstart,
      const int* __restrict__ tstart) {
  int lane = threadIdx.x & 31;
  int rt = blockIdx.x * 4 + (threadIdx.x >> 5);
  if (rt >= tstart[10]) return;
  int t = 0;
#pragma unroll
  for (int q = 1; q < 10; ++q)
    if (rt >= tstart[q]) t = q;
  int seg1 = rstart[t + 1];
  int slotbase = rstart[t] + (rt - tstart[t]) * 16;
  int rsub = lane & 15;
  int hi = lane >> 4;
  int klo = hi * 2;
  int slot = slotbase + rsub;
  int srow = perm[slot < seg1 ? slot : seg1 - 1];
  int K = K1 + K2;
  const float* Wt = W + (size_t)t * K * M;
  int cbase = blockIdx.y * 64;
  v8f z = {};
  v8f acc[4] = {z, z, z, z};
  wmma_panel(X + (size_t)srow * K1 + klo,
             Wt + (size_t)klo * M + cbase + rsub, K1 >> 2, M, acc);
  if (K2 > 0)
    wmma_panel(X2 + (size_t)srow * K2 + klo,
               Wt + (size_t)(K1 + klo) * M + cbase + rsub, K2 >> 2, M, acc);
  int prow[8]; bool pv[8];
#pragma unroll
  for (int v = 0; v < 8; ++v) {
    int sv = slotbase + v + 8 * hi;
    pv[v] = sv < seg1;
    prow[v] = perm[pv[v] ? sv : seg1 - 1];
  }
  const float* bt = Bv + (size_t)t * M;
#pragma unroll
  for (int j = 0; j < 4; ++j) {
    int col = cbase + j * 16 + rsub;
    float bv = bt[col];
#pragma unroll
    for (int v = 0; v < 8; ++v) {
      if (pv[v]) {
        float o = acc[j][v] + bv;
        if (act == 1) o = o > 0.f ? o : 0.f;
        else if (act == 2) o = o < 10.f ? o : 10.f;
        Y[(size_t)prow[v] * M + col] = o;
      }
    }
  }
}

// ---------------------------------------------------------------------------
// GATv2 edge passes (wave per edge; lanes 0-15 head0, 16-31 head1; 8 ch/lane)
// ---------------------------------------------------------------------------
__device__ __forceinline__ void atomicMaxF(float* addr, float val) {
  if (val >= 0.f) atomicMax((int*)addr, __float_as_int(val));
  else            atomicMin((unsigned int*)addr, __float_as_uint(val));
}

__global__ void __launch_bounds__(128)
gat_a_k(const float* __restrict__ xl, const float* __restrict__ xr,
        const int* __restrict__ src, const int* __restrict__ dst, int E, int N,
        const float* __restrict__ ea, const float* __restrict__ easum, float invE,
        const float* __restrict__ We, const float* __restrict__ att,
        float* __restrict__ a_out, float* __restrict__ amax) {
  int lane = threadIdx.x & 31;
  int e = blockIdx.x * 4 + (threadIdx.x >> 5);
  if (e >= E + N) return;
  int s, d;
  float eav[13];
  if (e < E) {
    s = src[e]; d = dst[e];
#pragma unroll
    for (int k = 0; k < 13; ++k) eav[k] = ea[(size_t)e * 13 + k];
  } else {
    s = d = e - E;
#pragma unroll
    for (int k = 0; k < 13; ++k) eav[k] = easum[k] * invE;
  }
  int cc0 = lane * 8;
  const float4* pl4 = (const float4*)(xl + (size_t)s * 256 + cc0);
  const float4* pr4 = (const float4*)(xr + (size_t)d * 256 + cc0);
  float4 l0 = pl4[0], l1 = pl4[1];
  float4 r0 = pr4[0], r1 = pr4[1];
  float lv[8] = {l0.x, l0.y, l0.z, l0.w, l1.x, l1.y, l1.z, l1.w};
  float rv[8] = {r0.x, r0.y, r0.z, r0.w, r1.x, r1.y, r1.z, r1.w};
  float acc = 0.f;
#pragma unroll
  for (int i = 0; i < 8; ++i) {
    int cc = cc0 + i;
    float ee = 0.f;
#pragma unroll
    for (int k = 0; k < 13; ++k) ee += eav[k] * We[k * 256 + cc];
    float m = lv[i] + rv[i] + ee;
    m = m > 0.f ? m : 0.2f * m;   // leaky relu 0.2
    acc += m * att[cc];
  }
#pragma unroll
  for (int off = 1; off < 16; off <<= 1) acc += __shfl_xor(acc, off, 32);
  if ((lane & 15) == 0) {
    int h = lane >> 4;
    a_out[(size_t)e * 2 + h] = acc;
    atomicMaxF(&amax[(size_t)d * 2 + h], acc);
  }
}

__global__ void gat_exp_k(const float* __restrict__ a, const int* __restrict__ dst,
                          int E, int N, const float* __restrict__ amax,
                          float* __restrict__ expa, float* __restrict__ asum) {
  int idx = blockIdx.x * blockDim.x + threadIdx.x;
  if (idx >= (E + N) * 2) return;
  int e = idx >> 1, h = idx & 1;
  int d = e < E ? dst[e] : e - E;
  float ex = __expf(a[idx] - amax[(size_t)d * 2 + h]);
  expa[idx] = ex;
  atomicAdd(&asum[(size_t)d * 2 + h], ex);
}

__global__ void __launch_bounds__(128)
gat_out_k(const float* __restrict__ xl, const int* __restrict__ src,
          const int* __restrict__ dst, int E, int N,
          const float* __restrict__ expa, const float* __restrict__ asum,
          float* __restrict__ out) {
  int lane = threadIdx.x & 31;
  int e = blockIdx.x * 4 + (threadIdx.x >> 5);
  if (e >= E + N) return;
  int s = e < E ? src[e] : e - E;
  int d = e < E ? dst[e] : e - E;
  int cc0 = lane * 8;
  int h = lane >> 4;
  float alpha = expa[(size_t)e * 2 + h] / (asum[(size_t)d * 2 + h] + 1e-16f);
  const float4* pl4 = (const float4*)(xl + (size_t)s * 256 + cc0);
  float4 l0 = pl4[0], l1 = pl4[1];
  float lv[8] = {l0.x, l0.y, l0.z, l0.w, l1.x, l1.y, l1.z, l1.w};
  float* po = out + (size_t)d * 256 + cc0;
#pragma unroll
  for (int i = 0; i < 8; ++i) atomicAdd(&po[i], alpha * lv[i]);
}

// ---------------------------------------------------------------------------
// host
// ---------------------------------------------------------------------------
static inline int cdiv(int a, int b) { return (a + b - 1) / b; }

extern "C" void kernel_launch(void* const* d_in, const int* in_sizes, int n_in,
                              void* d_out, int out_size, void* d_ws, size_t ws_size,
                              hipStream_t stream) {
  const int N = in_sizes[0] / 32;      // 60000
  const int E = in_sizes[2] / 13;      // 240000
  const int B = in_sizes[6] / 512;     // 2048
  const int EN = E + N;

  auto IN = [&](int i) { return (const float*)d_in[i]; };
  const float* x    = IN(0);
  const int* ei     = (const int*)d_in[1];
  const int* srcp   = ei;
  const int* dstp   = ei + E;
  const float* eaP  = IN(2);
  const int* types  = (const int*)d_in[3];
  const int* batchp = (const int*)d_in[4];
  const float* epsA = IN(5);
  const float* epsM = IN(6);

  float* out = (float*)d_out;
  float* mu_o    = out;
  float* lv_o    = out + (size_t)N * 256;
  float* molmu_o = out + (size_t)2 * N * 256;
  float* mollv_o = molmu_o + (size_t)B * 512;
  float* atomz_o = mollv_o + (size_t)B * 512;
  float* molz_o  = atomz_o + (size_t)N * 256;

  // workspace carve
  float* ws = (float*)d_ws;
  size_t off = 0;
  auto alloc = [&](size_t n) { float* p = ws + off; off += n; return p; };
  float* y128 = alloc((size_t)N * 128);
  float* ys   = alloc((size_t)N * 256);
  float* g    = alloc((size_t)N * 256);
  float* xl   = alloc((size_t)N * 256);
  float* xr   = alloc((size_t)N * 256);
  float* t1   = alloc((size_t)N * 256);
  float* hb2  = ys;   // overlay ys+g  -> N*512 (free by mol stage)
  float* hb1  = xl;   // overlay xl+xr -> N*512
  float* abuf = alloc((size_t)EN * 2);
  float* expa = alloc((size_t)EN * 2);
  float* amax = alloc((size_t)N * 2);
  float* asum = alloc((size_t)N * 2);
  float* easm = alloc(16);
  float* hmol = alloc((size_t)B * 512);
  float* mb1  = alloc((size_t)B * 512);
  float* mb2  = alloc((size_t)B * 512);
  int* ibase  = (int*)(ws + off);
  int* cnt    = ibase;
  int* rstart = ibase + 16;
  int* tstart = ibase + 32;
  int* cursor = ibase + 48;
  int* perm   = ibase + 64;

  auto fillf = [&](float* p, float v, int n) {
    fillf_k<<<cdiv(n, 256), 256, 0, stream>>>(p, v, n);
  };
  auto gemm = [&](const float* X, int K1, const float* X2, int K2, const float* W,
                  const float* b, float* Y, int rows, int M, int act) {
    dim3 gr(cdiv(rows, 64), M / 64);
    gemm_k<<<gr, 128, 0, stream>>>(X, K1, X2, K2, W, b, Y, rows, M, act);
  };
  auto het = [&](const float* X, int K1, const float* X2, int K2, const float* W,
                 const float* b, float* Y, int act) {
    int maxtiles = cdiv(N, 16) + 10;
    dim3 gr(cdiv(maxtiles, 4), 4);
    het_k<<<gr, 128, 0, stream>>>(X, K1, X2, K2, W, b, Y, 256, act, perm, rstart, tstart);
  };
  // param block for a GAT layer: pb: We, Wl, Wr, att, bias, bl, br
  auto run_gat = [&](const float* yin, int Kin, int pb, float* outb) {
    gemm(yin, Kin, nullptr, 0, IN(pb + 1), IN(pb + 5), xl, N, 256, 0);  // Wl,bl
    gemm(yin, Kin, nullptr, 0, IN(pb + 2), IN(pb + 6), xr, N, 256, 0);  // Wr,br
    fillf(amax, -INFINITY, N * 2);
    fillf(asum, 0.f, N * 2);
    biasb_k<<<cdiv(N * 256, 256), 256, 0, stream>>>(outb, IN(pb + 4), N * 256, 256);
    gat_a_k<<<cdiv(EN, 4), 128, 0, stream>>>(xl, xr, srcp, dstp, E, N, eaP, easm,
                                             1.f / (float)E, IN(pb), IN(pb + 3), abuf, amax);
    gat_exp_k<<<cdiv(EN * 2, 256), 256, 0, stream>>>(abuf, dstp, E, N, amax, expa, asum);
    gat_out_k<<<cdiv(EN, 4), 128, 0, stream>>>(xl, srcp, dstp, E, N, expa, asum, outb);
  };

  // ---- type bucketing (for hetero layers) + edge-attr mean ----
  filli_k<<<1, 64, 0, stream>>>(ibase, 0, 64);
  hist_k<<<cdiv(N, 256), 256, 0, stream>>>(types, N, cnt);
  scan_k<<<1, 1, 0, stream>>>(cnt, rstart, tstart);
  perm_k<<<cdiv(N, 256), 256, 0, stream>>>(types, N, rstart, cursor, perm);
  fillf(easm, 0.f, 16);
  easum_k<<<cdiv(E, 256), 256, 0, stream>>>(eaP, E, easm);

  // ---- atom-level pipeline ----
  // param indices (JAX pytree: dict keys sorted):
  // 7:emb.W 8:emb.b | gat1:9-15 | gat_lv:16-22 | gat_mu:23-29
  // lv1:30,31 lv2:32,33 | mol_mlp:34-39 | mol_mu:40-45 | mol_std:46-51
  // mu1:52,53 mu2:54,55 | shared:56,57
  gemm(x, 32, nullptr, 0, IN(7), IN(8), y128, N, 128, 0);        // emb (no relu: 1 layer)
  run_gat(y128, 128, 9, g);                                      // gat1 -> g[N,256]
  het(g, 256, nullptr, 0, IN(56), IN(57), ys, 0);                // shared -> ys
  run_gat(ys, 256, 23, g);                                       // gat_mu -> g
  het(g, 256, x, 32, IN(52), IN(53), t1, 0);                     // mu1(concat)
  het(t1, 256, nullptr, 0, IN(54), IN(55), mu_o, 0);             // mu2 -> out.mu
  run_gat(ys, 256, 16, g);                                       // gat_lv -> g
  het(g, 256, x, 32, IN(30), IN(31), t1, 0);                     // lv1(concat)
  het(t1, 256, nullptr, 0, IN(32), IN(33), lv_o, 2);             // lv2 + clamp(10)
  reparam_k<<<cdiv(N * 256, 256), 256, 0, stream>>>(mu_o, lv_o, epsA, atomz_o, N * 256);

  // ---- molecule-level pipeline ----
  gemm(atomz_o, 256, nullptr, 0, IN(34), IN(37), hb1, N, 512, 1);  // mol_mlp L0 relu
  gemm(hb1, 512, nullptr, 0, IN(35), IN(38), hb2, N, 512, 1);      // L1 relu
  gemm(hb2, 512, nullptr, 0, IN(36), IN(39), hb1, N, 512, 0);      // L2
  fillf(hmol, 0.f, B * 512);
  segsum_k<<<cdiv(N * 512, 256), 256, 0, stream>>>(hb1, batchp, hmol, N * 512);

  gemm(hmol, 512, nullptr, 0, IN(40), IN(43), mb1, B, 512, 1);     // mol_mu L0 relu
  gemm(mb1, 512, nullptr, 0, IN(41), IN(44), mb2, B, 512, 1);      // L1 relu
  gemm(mb2, 512, nullptr, 0, IN(42), IN(45), molmu_o, B, 512, 0);  // L2
  gemm(hmol, 512, nullptr, 0, IN(46), IN(49), mb1, B, 512, 1);     // mol_std L0 relu
  gemm(mb1, 512, nullptr, 0, IN(47), IN(50), mb2, B, 512, 1);      // L1 relu
  gemm(mb2, 512, nullptr, 0, IN(48), IN(51), mollv_o, B, 512, 0);  // L2
  reparam_k<<<cdiv(B * 512, 256), 256, 0, stream>>>(molmu_o, mollv_o, epsM, molz_o, B * 512);

  (void)n_in; (void)out_size; (void)ws_size;
}